// YoloLoss_14671608283137
// MI455X (gfx1250) — compile-verified
//
#include <hip/hip_runtime.h>

// YOLO-v1 loss on MI455X (gfx1250, wave32).
// Roofline: 2 x 802,816 x 120B = ~193 MB streamed once -> ~8.3 us at 23.3 TB/s;
// ~0.1 GFLOP of VALU. Pure HBM-bound: kernel 1 streams cell pairs with b128
// nontemporal loads; kernel 2 folds block partials 64-at-a-time with
// V_WMMA_F32_16X16X4_F32 (A = ones, exact f32 MAC, deterministic).

#define NBLOCKS  1024
#define NTHREADS 256

typedef __attribute__((ext_vector_type(2))) float v2f;
typedef __attribute__((ext_vector_type(4))) float v4f;
typedef __attribute__((ext_vector_type(8))) float v8f;

// Per-cell YOLO loss; p/t point at 30 consecutive floats (local arrays -> VGPRs).
__device__ __forceinline__ float cell_loss(const float* p, const float* t)
{
    const float t5     = t[5];
    const float fcoord = (t5 >  0.0f) ? 1.0f : 0.0f;   // object cell
    const float fnoobj = (t5 == 0.0f) ? 1.0f : 0.0f;   // no-object cell

    // confidence diffs (indices 4 and 9) -- used by both noobj and contain terms
    const float d4 = p[4] - t[4];
    const float d9 = p[9] - t[9];
    const float noobj_l = d4*d4 + d9*d9;

    // class loss over elements 10..29
    float cls = 0.0f;
    #pragma unroll
    for (int i = 10; i < 30; ++i) { float d = p[i] - t[i]; cls += d*d; }

    // pairwise IoU of 2 pred boxes vs 2 target boxes (corner format)
    float a1[2], a2[2];
    #pragma unroll
    for (int b = 0; b < 2; ++b) {
        a1[b] = (p[5*b+2] - p[5*b]) * (p[5*b+3] - p[5*b+1]);
        a2[b] = (t[5*b+2] - t[5*b]) * (t[5*b+3] - t[5*b+1]);
    }
    float iou[2][2];
    #pragma unroll
    for (int i = 0; i < 2; ++i) {
        #pragma unroll
        for (int j = 0; j < 2; ++j) {
            float ltx = fmaxf(p[5*i],     t[5*j]);
            float lty = fmaxf(p[5*i + 1], t[5*j + 1]);
            float rbx = fminf(p[5*i + 2], t[5*j + 2]);
            float rby = fminf(p[5*i + 3], t[5*j + 3]);
            float w = fmaxf(rbx - ltx, 0.0f);
            float h = fmaxf(rby - lty, 0.0f);
            float inter = w * h;
            iou[i][j] = inter / (a1[i] + a2[j] - inter + 1e-10f);
        }
    }
    // jnp.argmax picks the FIRST max on ties -> strict '>' for index 1
    const int m0 = (iou[1][0] > iou[0][0]) ? 1 : 0;
    const int m1 = (iou[1][1] > iou[0][1]) ? 1 : 0;
    const float r0 = ((m0 == 0) || (m1 == 0)) ? fcoord : 0.0f;
    const float r1 = ((m0 == 1) || (m1 == 1)) ? fcoord : 0.0f;

    // contain (confidence of responsible boxes)
    const float contain = r0 * d4 * d4 + r1 * d9 * d9;

    // localization (x1,y1,x2,y2 of responsible boxes)
    float loc = 0.0f;
    #pragma unroll
    for (int k = 0; k < 4; ++k) {
        float e0 = p[k]     - t[k];
        float e1 = p[5 + k] - t[5 + k];
        loc += r0 * e0 * e0 + r1 * e1 * e1;
    }

    return 5.0f * loc + contain + 0.5f * fnoobj * noobj_l + fcoord * cls;
}

__global__ __launch_bounds__(NTHREADS)
void yolo_cell_loss(const float* __restrict__ pred,
                    const float* __restrict__ tgt,
                    float* __restrict__ blocksums,
                    int ncells)
{
    float acc = 0.0f;
    const int gid    = blockIdx.x * blockDim.x + threadIdx.x;
    const int stride = gridDim.x * blockDim.x;

    // Pair of cells per thread: 60 floats = 240 B = 16B-aligned -> b128 loads.
    // Streamed-once data -> nontemporal (TH=NT) to keep it out of L2.
    const int npairs = ncells >> 1;
    for (int pair = gid; pair < npairs; pair += stride) {
        float P[60], T[60];
        const v4f* pp = reinterpret_cast<const v4f*>(pred) + (size_t)pair * 15;
        const v4f* tp = reinterpret_cast<const v4f*>(tgt)  + (size_t)pair * 15;
        #pragma unroll
        for (int i = 0; i < 15; ++i) {
            v4f a = __builtin_nontemporal_load(pp + i);
            P[4*i] = a.x; P[4*i+1] = a.y; P[4*i+2] = a.z; P[4*i+3] = a.w;
        }
        #pragma unroll
        for (int i = 0; i < 15; ++i) {
            v4f a = __builtin_nontemporal_load(tp + i);
            T[4*i] = a.x; T[4*i+1] = a.y; T[4*i+2] = a.z; T[4*i+3] = a.w;
        }
        acc += cell_loss(P, T);
        acc += cell_loss(P + 30, T + 30);
    }

    // odd tail cell (at most one thread takes this)
    for (int cell = 2 * npairs + gid; cell < ncells; cell += stride) {
        float P[30], T[30];
        const float2* pp = reinterpret_cast<const float2*>(pred) + (size_t)cell * 15;
        const float2* tp = reinterpret_cast<const float2*>(tgt)  + (size_t)cell * 15;
        #pragma unroll
        for (int i = 0; i < 15; ++i) { float2 a = pp[i]; P[2*i] = a.x; P[2*i+1] = a.y; }
        #pragma unroll
        for (int i = 0; i < 15; ++i) { float2 a = tp[i]; T[2*i] = a.x; T[2*i+1] = a.y; }
        acc += cell_loss(P, T);
    }

    // deterministic wave32 reduction
    #pragma unroll
    for (int off = 16; off > 0; off >>= 1)
        acc += __shfl_down(acc, off, 32);

    __shared__ float wsum[NTHREADS / 32];
    const int lane = threadIdx.x & 31;
    const int wave = threadIdx.x >> 5;
    if (lane == 0) wsum[wave] = acc;
    __syncthreads();
    if (threadIdx.x == 0) {
        float b = 0.0f;
        #pragma unroll
        for (int i = 0; i < NTHREADS / 32; ++i) b += wsum[i];
        blocksums[blockIdx.x] = b;
    }
}

// Final reduction: one wave, V_WMMA_F32_16X16X4_F32 with A = ones.
// D[m][n] += sum_k B[k][n]; each WMMA folds 64 partials. Since A is all ones,
// ANY bijection of the 64-chunk into B's slots yields the same total, so lane L
// simply loads chunk[2L], chunk[2L+1] as one float2 (coalesced b64, no branches).
// n MUST be a multiple of 64 (guaranteed by the launcher).
__global__ __launch_bounds__(32)
void yolo_final_reduce(const float* __restrict__ part, float* __restrict__ out, int n)
{
    const int lane = threadIdx.x;
    v8f c = {};
    v2f a; a[0] = 1.0f; a[1] = 1.0f;     // all-ones A (16x4 f32, 2 VGPRs/lane)

    const float2* p2 = reinterpret_cast<const float2*>(part) + lane;
    const int nchunks = n >> 6;
    for (int ch = 0; ch < nchunks; ++ch) {
        float2 x = p2[ch * 32];
        v2f b; b[0] = x.x; b[1] = x.y;
        c = __builtin_amdgcn_wmma_f32_16x16x4_f32(
                /*neg_a=*/false, a, /*neg_b=*/false, b,
                /*c_mod=*/(short)0, c, /*reuse_a=*/false, /*reuse_b=*/false);
    }

    // c[0]: lanes 0..15 hold the 16 column sums (rows of D identical, A = ones)
    float v = c[0];
    #pragma unroll
    for (int off = 8; off > 0; off >>= 1)
        v += __shfl_down(v, off, 16);
    if (lane == 0) out[0] = v;
}

extern "C" void kernel_launch(void* const* d_in, const int* in_sizes, int n_in,
                              void* d_out, int out_size, void* d_ws, size_t ws_size,
                              hipStream_t stream) {
    const float* pred = (const float*)d_in[0];   // [batch, 7*7*30] f32
    const float* tgt  = (const float*)d_in[1];   // [batch, 7, 7, 30] f32
    float* out   = (float*)d_out;                // scalar loss
    float* bsums = (float*)d_ws;

    const int ncells = in_sizes[1] / 30;         // batch * 49

    // nblocks: multiple of 64 so kernel 2 needs no bounds checks; fits in d_ws.
    int nblocks = NBLOCKS;
    if (ws_size < (size_t)nblocks * sizeof(float)) {
        nblocks = (int)((ws_size / sizeof(float)) & ~(size_t)63);
        if (nblocks < 64) nblocks = 64;          // d_ws is always >= 256 B in practice
    }

    yolo_cell_loss<<<nblocks, NTHREADS, 0, stream>>>(pred, tgt, bsums, ncells);
    yolo_final_reduce<<<1, 32, 0, stream>>>(bsums, out, nblocks);
}